// LearnableMask_19963007991897
// MI455X (gfx1250) — compile-verified
//
#include <hip/hip_runtime.h>
#include <hip/hip_bf16.h>

// Problem constants (fixed by the reference setup).
#define BB 64
#define NN 4096
#define DD 256
#define KK 1024

typedef __attribute__((ext_vector_type(2))) float v2f;
typedef __attribute__((ext_vector_type(8))) float v8f;

// ---------------------------------------------------------------------------
// Kernel 1: logits[b,n] = dot(x[b,n,:], W) + bias  via V_WMMA_F32_16X16X4_F32.
// Each wave owns a tile of 16 consecutive patches (flattened over B*N; N is a
// multiple of 16 so tiles never straddle batch rows). B-matrix = W broadcast
// to all 16 columns, so every column of D holds the same logit.
//
// A 16x4 f32 layout (ISA 7.12.2): lanes 0-15 -> M=0..15, VGPR0 = K {0 | 2}
// (low | high half-wave), VGPR1 = K {1 | 3}.  C/D: VGPR j, half h -> M = j+8h.
// ---------------------------------------------------------------------------
__global__ void __launch_bounds__(256)
logits_wmma_kernel(const float* __restrict__ x, const float* __restrict__ W,
                   const float* __restrict__ bias, float* __restrict__ logits)
{
    const int lane = threadIdx.x & 31;
    const int wave = threadIdx.x >> 5;
    const long tile = (long)blockIdx.x * (blockDim.x >> 5) + wave;  // 16 rows per tile
    const long row0 = tile * 16;                                    // flattened B*N row
    const int m    = lane & 15;
    const int half = lane >> 4;

    const float* xr = x + (row0 + m) * DD + half * 2;
    v8f acc = {};
    #pragma unroll 8
    for (int k = 0; k < DD; k += 4) {
        v2f a, b;
        a.x = xr[k];
        a.y = xr[k + 1];
        b.x = W[k + half * 2];
        b.y = W[k + half * 2 + 1];
        // 8 args: (neg_a, A, neg_b, B, c_mod, C, reuse_a, reuse_b)
        acc = __builtin_amdgcn_wmma_f32_16x16x4_f32(
            false, a, false, b, (short)0, acc, false, false);
    }
    // All 16 columns are identical; lane 0 / lane 16 extract M=0..7 / M=8..15.
    if ((lane & 15) == 0) {
        const float bv = bias[0];
        #pragma unroll
        for (int j = 0; j < 8; ++j)
            logits[row0 + half * 8 + j] = acc[j] + bv;
    }
}

// ---------------------------------------------------------------------------
// Kernel 2: per batch row -- softmax over N=4096, full descending bitonic
// argsort of (prob, idx) in LDS, then scatter ids_shuffle / hard_mask /
// ids_restore.  One 1024-thread block per row; 36 KB LDS.
// ---------------------------------------------------------------------------
__global__ void __launch_bounds__(1024)
softmax_sort_kernel(const float* __restrict__ logits, float* __restrict__ probs,
                    int* __restrict__ ids_shuffle, float* __restrict__ out_mask,
                    float* __restrict__ out_restore)
{
    __shared__ float key[NN];
    __shared__ int   val[NN];
    __shared__ float red[1024];

    const int  t   = threadIdx.x;
    const long row = (long)blockIdx.x * NN;

    // Load logits, track local max.
    float lmax = -3.402823466e38f;
    for (int i = t; i < NN; i += 1024) {
        float v = logits[row + i];
        key[i] = v;
        val[i] = i;
        lmax = fmaxf(lmax, v);
    }
    red[t] = lmax;
    __syncthreads();
    for (int s = 512; s > 0; s >>= 1) {
        if (t < s) red[t] = fmaxf(red[t], red[t + s]);
        __syncthreads();
    }
    const float mx = red[0];
    __syncthreads();

    // exp + sum
    float lsum = 0.0f;
    for (int i = t; i < NN; i += 1024) {
        float e = expf(key[i] - mx);
        key[i] = e;
        lsum += e;
    }
    red[t] = lsum;
    __syncthreads();
    for (int s = 512; s > 0; s >>= 1) {
        if (t < s) red[t] += red[t + s];
        __syncthreads();
    }
    const float inv = 1.0f / red[0];
    __syncthreads();

    for (int i = t; i < NN; i += 1024) {
        float p = key[i] * inv;
        key[i] = p;
        probs[row + i] = p;
    }
    __syncthreads();

    // Bitonic sort: descending by prob, ties broken by ascending index
    // (matches jax stable argsort of -probs / top_k ordering).
    for (int kk = 2; kk <= NN; kk <<= 1) {
        for (int j = kk >> 1; j > 0; j >>= 1) {
            for (int i = t; i < NN; i += 1024) {
                const int ixj = i ^ j;
                if (ixj > i) {
                    const bool up = ((i & kk) == 0);   // "up" = our descending order
                    float pa = key[i], pb = key[ixj];
                    int   ia = val[i], ib = val[ixj];
                    const bool aFirst = (pa > pb) || (pa == pb && ia < ib);
                    if (up ? !aFirst : aFirst) {
                        key[i] = pb; key[ixj] = pa;
                        val[i] = ib; val[ixj] = ia;
                    }
                }
            }
            __syncthreads();
        }
    }

    // Scatter outputs. val[] is a permutation of [0,N): every slot written.
    for (int i = t; i < NN; i += 1024) {
        const int idx = val[i];
        ids_shuffle[row + i]    = idx;
        out_restore[row + idx]  = (float)i;                 // ids_restore
        out_mask[row + idx]     = (i < KK) ? 0.0f : 1.0f;   // hard_mask
    }
}

// ---------------------------------------------------------------------------
// Kernel 3: x_masked[b,k,:] = x[b, topk_idx[b,k], :] * probs[b, topk_idx[b,k]]
// 4 (b,k) rows per 256-thread block, float4 fully-coalesced.
// ---------------------------------------------------------------------------
__global__ void __launch_bounds__(256)
gather_scale_kernel(const float* __restrict__ x, const float* __restrict__ probs,
                    const int* __restrict__ ids_shuffle, float* __restrict__ out)
{
    const int t  = threadIdx.x;
    const int r  = t >> 6;   // row within block: 0..3
    const int d4 = t & 63;   // float4 index:     0..63
    const long bk = (long)blockIdx.x * 4 + r;   // 0 .. B*K-1
    const int b = (int)(bk >> 10);              // / K
    const int k = (int)(bk & (KK - 1));

    const int   idx = ids_shuffle[(long)b * NN + k];
    const float p   = probs[(long)b * NN + idx];

    const float4* src = (const float4*)(x + ((long)b * NN + idx) * DD);
    float4 v = src[d4];
    v.x *= p; v.y *= p; v.z *= p; v.w *= p;
    ((float4*)(out + bk * DD))[d4] = v;
}

// ---------------------------------------------------------------------------
extern "C" void kernel_launch(void* const* d_in, const int* in_sizes, int n_in,
                              void* d_out, int out_size, void* d_ws, size_t ws_size,
                              hipStream_t stream) {
    const float* x    = (const float*)d_in[0];   // [B,N,D] f32
    const float* W    = (const float*)d_in[1];   // [1,D]   f32
    const float* bias = (const float*)d_in[2];   // [1]     f32
    // d_in[3] = K (device int); fixed at 1024 by the reference setup.

    float* out = (float*)d_out;
    float* out_masked  = out;                                  // [B,K,D]
    float* out_mask    = out + (size_t)BB * KK * DD;           // [B,N]
    float* out_restore = out_mask + (size_t)BB * NN;           // [B,N] (as f32)

    // Workspace: [0, B*N) f32 logits->probs (in place), then [B*N) i32 ids.
    float* probs = (float*)d_ws;
    int*   ids   = (int*)((char*)d_ws + (size_t)BB * NN * sizeof(float));

    // 1) logits via WMMA: B*N/16 tiles, 8 waves (tiles) per 256-thread block.
    const int tiles  = (BB * NN) / 16;       // 16384
    const int blocks = tiles / 8;            // 2048
    logits_wmma_kernel<<<blocks, 256, 0, stream>>>(x, W, bias, probs);

    // 2) softmax + full descending argsort + mask/restore scatter, 1 block/row.
    softmax_sort_kernel<<<BB, 1024, 0, stream>>>(probs, probs, ids,
                                                 out_mask, out_restore);

    // 3) gather kept patches scaled by prob (straight-through forward value).
    gather_scale_kernel<<<(BB * KK) / 4, 256, 0, stream>>>(x, probs, ids,
                                                           out_masked);
}